// SpeciesGAT_6932077216414
// MI455X (gfx1250) — compile-verified
//
#include <hip/hip_runtime.h>
#include <math.h>

typedef __attribute__((ext_vector_type(2))) float v2f;
typedef __attribute__((ext_vector_type(8))) float v8f;

#define IN_DIM 128
#define NEG_SLOPE 0.2f

// ---- monotone float<->uint encoding so atomicMax(uint) == float max ----
__device__ __forceinline__ unsigned encf(float f) {
  unsigned u = __float_as_uint(f);
  return (u & 0x80000000u) ? ~u : (u | 0x80000000u);
}
__device__ __forceinline__ float decf(unsigned u) {
  return (u & 0x80000000u) ? __uint_as_float(u & 0x7FFFFFFFu) : __uint_as_float(~u);
}
__device__ __forceinline__ float lrelu(float v) { return v > 0.f ? v : NEG_SLOPE * v; }

__device__ __forceinline__ void edge_sd(const int* __restrict__ ei, int E, int e,
                                        int& s, int& d) {
  if (e < E) { s = ei[e]; d = ei[E + e]; }
  else       { s = e - E; d = e - E; }       // appended self-loops
}

// ---------------- init: zero accumulators, -inf (encoded) maxima ----------------
__global__ void k_init(float* __restrict__ agg1, unsigned* __restrict__ m1,
                       float* __restrict__ den1, unsigned* __restrict__ m2,
                       float* __restrict__ den2, float* __restrict__ agg2, int nNodes) {
  int i = blockIdx.x * blockDim.x + threadIdx.x;
  if (i < nNodes * IN_DIM) agg1[i] = 0.f;
  if (i < nNodes * 2) { m1[i] = 0x007FFFFFu; den1[i] = 0.f; }   // enc(-inf)
  if (i < nNodes)     { m2[i] = 0x007FFFFFu; den2[i] = 0.f; agg2[i] = 0.f; }
}

// ---------------- GEMM1: H = X[ N x128 ] * W1[128 x128 ], fp32 WMMA ----------------
// One wave computes a 16x16 tile; K-loop of 32 x V_WMMA_F32_16X16X4_F32.
__global__ void k_gemm1_wmma(const float* __restrict__ X, const float* __restrict__ W,
                             float* __restrict__ H, int nNodes) {
  int wave = (blockIdx.x * blockDim.x + threadIdx.x) >> 5;
  int lane = threadIdx.x & 31;
  int rowTiles = (nNodes + 15) >> 4;
  int rt = wave >> 3;          // 128/16 = 8 column tiles
  int ct = wave & 7;
  if (rt >= rowTiles) return;  // wave-uniform: EXEC stays all-ones for WMMA
  int row0 = rt * 16, col0 = ct * 16;
  int half = lane >> 4, idx = lane & 15;
  int arow = row0 + idx; if (arow > nNodes - 1) arow = nNodes - 1;  // clamp, keep EXEC full
  const float* xrow = X + (size_t)arow * IN_DIM;

  v8f c = {};
  for (int k0 = 0; k0 < IN_DIM; k0 += 4) {
    int k = k0 + 2 * half;     // ISA 16x4 f32 A layout: lanes0-15 K={k0,k0+1}, lanes16-31 K={k0+2,k0+3}
    v2f a, b;
    a[0] = xrow[k];
    a[1] = xrow[k + 1];
    b[0] = W[(size_t)k * IN_DIM + col0 + idx];        // B 4x16: N across lanes, same K split
    b[1] = W[(size_t)(k + 1) * IN_DIM + col0 + idx];
    c = __builtin_amdgcn_wmma_f32_16x16x4_f32(false, a, false, b, (short)0, c, false, false);
  }
  // D layout: VGPR j -> row (j + 8*half), col idx.
  // Fast path: whole tile in range (wave-uniform check) -> unguarded full-EXEC stores.
  float* hbase = H + (size_t)row0 * IN_DIM + col0 + idx + (size_t)(8 * half) * IN_DIM;
  if (row0 + 16 <= nNodes) {
#pragma unroll
    for (int j = 0; j < 8; ++j) hbase[(size_t)j * IN_DIM] = c[j];
  } else {
#pragma unroll
    for (int j = 0; j < 8; ++j) {
      if (row0 + j + 8 * half < nNodes) hbase[(size_t)j * IN_DIM] = c[j];
    }
  }
}

// ---------------- per-node attention coefficients (heads=2, C=64) ----------------
// att_src1[2,64] flattened == channel index 0..127, so flat dot per head group.
__global__ void k_attn_coef1(const float* __restrict__ H, const float* __restrict__ attS,
                             const float* __restrict__ attD, float* __restrict__ aS,
                             float* __restrict__ aD, int nNodes) {
  int wave = (blockIdx.x * blockDim.x + threadIdx.x) >> 5;
  int lane = threadIdx.x & 31;
  if (wave >= nNodes) return;
  const float* hr = H + (size_t)wave * IN_DIM + lane * 4;
  float s = 0.f, d = 0.f;
#pragma unroll
  for (int j = 0; j < 4; ++j) {
    float hv = hr[j];
    s += hv * attS[lane * 4 + j];
    d += hv * attD[lane * 4 + j];
  }
  // reduce within each 16-lane group (lanes 0-15 = head0, 16-31 = head1)
  for (int m = 1; m < 16; m <<= 1) { s += __shfl_xor(s, m, 32); d += __shfl_xor(d, m, 32); }
  if ((lane & 15) == 0) {
    aS[wave * 2 + (lane >> 4)] = s;
    aD[wave * 2 + (lane >> 4)] = d;
  }
}

// ---------------- layer-1 edge passes ----------------
__global__ void k_edge_max1(const int* __restrict__ ei, int E, int Etot,
                            const float* __restrict__ aS, const float* __restrict__ aD,
                            unsigned* __restrict__ m1) {
  int e = blockIdx.x * blockDim.x + threadIdx.x;
  if (e >= Etot) return;
  int s, d; edge_sd(ei, E, e, s, d);
#pragma unroll
  for (int h = 0; h < 2; ++h) {
    float v = lrelu(aS[s * 2 + h] + aD[d * 2 + h]);
    atomicMax(&m1[d * 2 + h], encf(v));
  }
}

__global__ void k_edge_denom1(const int* __restrict__ ei, int E, int Etot,
                              const float* __restrict__ aS, const float* __restrict__ aD,
                              const unsigned* __restrict__ m1, float* __restrict__ den1) {
  int e = blockIdx.x * blockDim.x + threadIdx.x;
  if (e >= Etot) return;
  int s, d; edge_sd(ei, E, e, s, d);
#pragma unroll
  for (int h = 0; h < 2; ++h) {
    float v = lrelu(aS[s * 2 + h] + aD[d * 2 + h]);
    atomicAdd(&den1[d * 2 + h], expf(v - decf(m1[d * 2 + h])));
  }
}

// wave per edge: 32 lanes x float4 = 128 channels; head = lane>>4
__global__ void k_edge_agg1(const int* __restrict__ ei, int E, int Etot,
                            const float* __restrict__ aS, const float* __restrict__ aD,
                            const unsigned* __restrict__ m1, const float* __restrict__ den1,
                            const float* __restrict__ H, float* __restrict__ agg) {
  int wave = (blockIdx.x * blockDim.x + threadIdx.x) >> 5;
  int lane = threadIdx.x & 31;
  if (wave >= Etot) return;
  int s, d; edge_sd(ei, E, wave, s, d);
  int h = lane >> 4;
  float v = lrelu(aS[s * 2 + h] + aD[d * 2 + h]);
  float alpha = expf(v - decf(m1[d * 2 + h])) / den1[d * 2 + h];
  int c0 = lane * 4;
  const float* hs = H + (size_t)s * IN_DIM + c0;
  float* od = agg + (size_t)d * IN_DIM + c0;
#pragma unroll
  for (int j = 0; j < 4; ++j) atomicAdd(&od[j], alpha * hs[j]);
}

// ---------------- finalize layer 1 (relu+bias, in place) + layer-2 mat-vec ----------------
__global__ void k_finalize1(float* __restrict__ agg, const float* __restrict__ b1,
                            const float* __restrict__ W2, float* __restrict__ z, int nNodes) {
  int wave = (blockIdx.x * blockDim.x + threadIdx.x) >> 5;
  int lane = threadIdx.x & 31;
  if (wave >= nNodes) return;
  float* row = agg + (size_t)wave * IN_DIM + lane * 4;
  float p = 0.f;
#pragma unroll
  for (int j = 0; j < 4; ++j) {
    float v = row[j] + b1[lane * 4 + j];
    v = v > 0.f ? v : 0.f;
    row[j] = v;                 // h2 stored in place
    p += v * W2[lane * 4 + j];
  }
  for (int m = 1; m < 32; m <<= 1) p += __shfl_xor(p, m, 32);
  if (lane == 0) z[wave] = p;
}

// ---------------- layer-2 edge passes (H=1, C=1) ----------------
__global__ void k_edge_max2(const int* __restrict__ ei, int E, int Etot,
                            const float* __restrict__ z, const float* __restrict__ as2,
                            const float* __restrict__ ad2, unsigned* __restrict__ m2) {
  int e = blockIdx.x * blockDim.x + threadIdx.x;
  if (e >= Etot) return;
  int s, d; edge_sd(ei, E, e, s, d);
  float v = lrelu(as2[0] * z[s] + ad2[0] * z[d]);
  atomicMax(&m2[d], encf(v));
}

__global__ void k_edge_denom2(const int* __restrict__ ei, int E, int Etot,
                              const float* __restrict__ z, const float* __restrict__ as2,
                              const float* __restrict__ ad2, const unsigned* __restrict__ m2,
                              float* __restrict__ den2) {
  int e = blockIdx.x * blockDim.x + threadIdx.x;
  if (e >= Etot) return;
  int s, d; edge_sd(ei, E, e, s, d);
  float v = lrelu(as2[0] * z[s] + ad2[0] * z[d]);
  atomicAdd(&den2[d], expf(v - decf(m2[d])));
}

__global__ void k_edge_agg2(const int* __restrict__ ei, int E, int Etot,
                            const float* __restrict__ z, const float* __restrict__ as2,
                            const float* __restrict__ ad2, const unsigned* __restrict__ m2,
                            const float* __restrict__ den2, float* __restrict__ agg2) {
  int e = blockIdx.x * blockDim.x + threadIdx.x;
  if (e >= Etot) return;
  int s, d; edge_sd(ei, E, e, s, d);
  float v = lrelu(as2[0] * z[s] + ad2[0] * z[d]);
  float alpha = expf(v - decf(m2[d])) / den2[d];
  atomicAdd(&agg2[d], alpha * z[s]);
}

__global__ void k_final_out(const float* __restrict__ agg2, const float* __restrict__ b2,
                            float* __restrict__ out, int nNodes) {
  int n = blockIdx.x * blockDim.x + threadIdx.x;
  if (n >= nNodes) return;
  float v = agg2[n] + b2[0];
  out[n] = 1.f / (1.f + expf(-v));
}

// ------------------------------------------------------------------
extern "C" void kernel_launch(void* const* d_in, const int* in_sizes, int n_in,
                              void* d_out, int out_size, void* d_ws, size_t ws_size,
                              hipStream_t stream) {
  const float* x     = (const float*)d_in[0];
  const int*   ei    = (const int*)  d_in[1];
  const float* W1    = (const float*)d_in[2];
  const float* attS1 = (const float*)d_in[3];
  const float* attD1 = (const float*)d_in[4];
  const float* b1    = (const float*)d_in[5];
  const float* W2    = (const float*)d_in[6];
  const float* as2   = (const float*)d_in[7];
  const float* ad2   = (const float*)d_in[8];
  const float* b2    = (const float*)d_in[9];
  float* out = (float*)d_out;

  const int nNodes = in_sizes[0] / IN_DIM;
  const int E      = in_sizes[1] / 2;
  const int Etot   = E + nNodes;     // reference appends self-loops

  // workspace carve-out (~107 MB)
  char* ws = (char*)d_ws;
  size_t off = 0;
  auto carve = [&](size_t bytes) -> char* {
    char* p = ws + off;
    off += (bytes + 255) & ~(size_t)255;
    return p;
  };
  float*    h1   = (float*)   carve((size_t)nNodes * IN_DIM * 4);
  float*    agg1 = (float*)   carve((size_t)nNodes * IN_DIM * 4);
  float*    aS   = (float*)   carve((size_t)nNodes * 2 * 4);
  float*    aD   = (float*)   carve((size_t)nNodes * 2 * 4);
  unsigned* m1   = (unsigned*)carve((size_t)nNodes * 2 * 4);
  float*    den1 = (float*)   carve((size_t)nNodes * 2 * 4);
  float*    z    = (float*)   carve((size_t)nNodes * 4);
  unsigned* m2   = (unsigned*)carve((size_t)nNodes * 4);
  float*    den2 = (float*)   carve((size_t)nNodes * 4);
  float*    agg2 = (float*)   carve((size_t)nNodes * 4);

  const int B = 256;
  int totInit = nNodes * IN_DIM;
  k_init<<<(totInit + B - 1) / B, B, 0, stream>>>(agg1, m1, den1, m2, den2, agg2, nNodes);

  int waves = ((nNodes + 15) / 16) * 8;               // 16x16 tiles, 8 col tiles
  k_gemm1_wmma<<<(waves * 32 + 127) / 128, 128, 0, stream>>>(x, W1, h1, nNodes);

  k_attn_coef1<<<(nNodes * 32 + B - 1) / B, B, 0, stream>>>(h1, attS1, attD1, aS, aD, nNodes);

  k_edge_max1  <<<(Etot + B - 1) / B, B, 0, stream>>>(ei, E, Etot, aS, aD, m1);
  k_edge_denom1<<<(Etot + B - 1) / B, B, 0, stream>>>(ei, E, Etot, aS, aD, m1, den1);
  k_edge_agg1  <<<((long long)Etot * 32 + B - 1) / B, B, 0, stream>>>(ei, E, Etot, aS, aD, m1, den1, h1, agg1);

  k_finalize1<<<(nNodes * 32 + B - 1) / B, B, 0, stream>>>(agg1, b1, W2, z, nNodes);

  k_edge_max2  <<<(Etot + B - 1) / B, B, 0, stream>>>(ei, E, Etot, z, as2, ad2, m2);
  k_edge_denom2<<<(Etot + B - 1) / B, B, 0, stream>>>(ei, E, Etot, z, as2, ad2, m2, den2);
  k_edge_agg2  <<<(Etot + B - 1) / B, B, 0, stream>>>(ei, E, Etot, z, as2, ad2, m2, den2, agg2);

  k_final_out<<<(nNodes + B - 1) / B, B, 0, stream>>>(agg2, b2, out, nNodes);
}